// AdaptiveCosineNCC_1743756722460
// MI455X (gfx1250) — compile-verified
//
#include <hip/hip_runtime.h>
#include <math.h>

#define D     512
#define NWAY  64
#define NQ    65536
#define NS    65536
#define EPSF  1e-30f
#define MBLK  128
#define KC    64
#define LDSTR 68   // 64 + 4 floats pad: bank = (4*row + k) % 64, conflict-free; 272B rows stay 16B-aligned

typedef float v2f __attribute__((ext_vector_type(2)));
typedef float v4f __attribute__((ext_vector_type(4)));
typedef float v8f __attribute__((ext_vector_type(8)));

// ---------------------------------------------------------------------------
// Pass 1: deterministic per-class partial sums.
// grid = 128 blocks (64 row-chunks x 2 dim-halves), 256 threads.
// Each LDS cell (class, dim) has exactly one writer thread, rows in fixed order.
// ---------------------------------------------------------------------------
__global__ void __launch_bounds__(256)
proto_partial(const float* __restrict__ sup, const int* __restrict__ lab,
              float* __restrict__ pSums) {
  __shared__ float ls[NWAY * 256];
  const int t    = threadIdx.x;
  const int bx   = blockIdx.x;
  const int rc   = bx >> 1;
  const int half = bx & 1;
  for (int i = t; i < NWAY * 256; i += 256) ls[i] = 0.f;
  __syncthreads();
  const int base = rc * 1024;
  const float* src = sup + (size_t)base * D + half * 256 + t;
  for (int r = 0; r < 1024; ++r) {
    int L = lab[base + r];                 // uniform across wave -> scalar load
    ls[L * 256 + t] += src[(size_t)r * D]; // coalesced 1KB row segment
  }
  __syncthreads();
  float* dst = pSums + (size_t)bx * (NWAY * 256);
  for (int i = t; i < NWAY * 256; i += 256) dst[i] = ls[i];
}

// ---------------------------------------------------------------------------
// Pass 2: finalize prototypes. grid = 64 (one class per block), 256 threads.
// count by label scan; reduce 128 partials in fixed order; fold scale/||p|| in.
// ---------------------------------------------------------------------------
__global__ void __launch_bounds__(256)
proto_final(const float* __restrict__ pSums, const int* __restrict__ lab,
            const float* __restrict__ scaleP, float* __restrict__ protN) {
  __shared__ float red[256];
  __shared__ int   redi[256];
  const int c = blockIdx.x;
  const int t = threadIdx.x;

  int cnt = 0;
  for (int i = t; i < NS; i += 256) cnt += (lab[i] == c) ? 1 : 0;
  redi[t] = cnt;
  __syncthreads();
  for (int s = 128; s > 0; s >>= 1) {
    if (t < s) redi[t] += redi[t + s];
    __syncthreads();
  }
  const float fcnt = (float)redi[0];

  float p[2];
  #pragma unroll
  for (int j = 0; j < 2; ++j) {
    int d = t + j * 256;
    int half = d >> 8, dcol = d & 255;
    float s = 0.f;
    for (int rcb = 0; rcb < 64; ++rcb)   // fixed order -> deterministic
      s += pSums[(size_t)((rcb * 2 + half) * NWAY + c) * 256 + dcol];
    p[j] = s / fcnt;
  }

  red[t] = p[0] * p[0] + p[1] * p[1];
  __syncthreads();
  for (int s = 128; s > 0; s >>= 1) {
    if (t < s) red[t] += red[t + s];
    __syncthreads();
  }
  const float fac = scaleP[0] / fmaxf(sqrtf(red[0]), EPSF);
  protN[(size_t)c * D + t]       = p[0] * fac;
  protN[(size_t)c * D + t + 256] = p[1] * fac;
}

// ---------------------------------------------------------------------------
// Fused cosine GEMM: logits[q][c] = (query[q] . protN[c]) / max(||query[q]||,eps)
// grid = 512 blocks of 128 queries, 256 threads (8 waves).
// Wave w: 16 queries x 64 classes via 4x V_WMMA_F32_16X16X4_F32 accumulators.
// Query norms computed from the same LDS-staged data (no extra HBM pass).
// ---------------------------------------------------------------------------
__global__ void __launch_bounds__(256)
cosine_gemm(const float* __restrict__ q, const float* __restrict__ protN,
            float* __restrict__ out) {
  __shared__ float qs[MBLK * LDSTR];
  __shared__ float ps[NWAY * LDSTR];
  __shared__ float normP[256];
  __shared__ float invq[MBLK];

  const int t    = threadIdx.x;
  const int w    = t >> 5;      // wave id 0..7
  const int lane = t & 31;
  const int l15  = lane & 15;
  const int hi   = lane >> 4;   // 0: lanes 0-15, 1: lanes 16-31
  const size_t qBase = (size_t)blockIdx.x * MBLK;

  v8f acc0 = {}, acc1 = {}, acc2 = {}, acc3 = {};
  float nacc = 0.f;
  const int nrow  = t & 127;
  const int nhalf = t >> 7;

  for (int kc = 0; kc < D; kc += KC) {
    // stage query chunk: 128 rows x 64 k -> 2048 float4, 8 per thread
    #pragma unroll
    for (int j = 0; j < 8; ++j) {
      int v = t + j * 256;
      int row = v >> 4, c4 = (v & 15) * 4;
      v4f val = *(const v4f*)(q + (qBase + row) * D + kc + c4);
      *(v4f*)(qs + row * LDSTR + c4) = val;
    }
    // stage prototype chunk: 64 rows x 64 k -> 1024 float4, 4 per thread
    #pragma unroll
    for (int j = 0; j < 4; ++j) {
      int v = t + j * 256;
      int row = v >> 4, c4 = (v & 15) * 4;
      v4f val = *(const v4f*)(protN + (size_t)row * D + kc + c4);
      *(v4f*)(ps + row * LDSTR + c4) = val;
    }
    __syncthreads();

    // query-norm partials from LDS (fixed per-thread order -> deterministic)
    #pragma unroll
    for (int i = 0; i < 8; ++i) {
      v4f x = *(const v4f*)(qs + nrow * LDSTR + nhalf * 32 + i * 4);
      nacc += x.x * x.x + x.y * x.y + x.z * x.z + x.w * x.w;
    }

    // WMMA over this K chunk. f32 A 16x4 layout: lanes0-15 K={k,k+1},
    // lanes16-31 K={k+2,k+3}; B mirrors with class rows.
    for (int k4 = 0; k4 < KC; k4 += 4) {
      v2f a  = *(const v2f*)(qs + (w * 16 + l15) * LDSTR + k4 + 2 * hi);
      v2f b0 = *(const v2f*)(ps + ( 0 + l15) * LDSTR + k4 + 2 * hi);
      v2f b1 = *(const v2f*)(ps + (16 + l15) * LDSTR + k4 + 2 * hi);
      v2f b2 = *(const v2f*)(ps + (32 + l15) * LDSTR + k4 + 2 * hi);
      v2f b3 = *(const v2f*)(ps + (48 + l15) * LDSTR + k4 + 2 * hi);
      acc0 = __builtin_amdgcn_wmma_f32_16x16x4_f32(false, a, false, b0, (short)0, acc0, false, false);
      acc1 = __builtin_amdgcn_wmma_f32_16x16x4_f32(false, a, false, b1, (short)0, acc1, false, false);
      acc2 = __builtin_amdgcn_wmma_f32_16x16x4_f32(false, a, false, b2, (short)0, acc2, false, false);
      acc3 = __builtin_amdgcn_wmma_f32_16x16x4_f32(false, a, false, b3, (short)0, acc3, false, false);
    }
    __syncthreads();
  }

  // finalize 1/||q|| (deterministic pairwise combine)
  normP[t] = nacc;
  __syncthreads();
  if (t < MBLK) {
    float n2 = normP[t] + normP[t + MBLK];
    invq[t] = 1.f / fmaxf(sqrtf(n2), EPSF);
  }
  __syncthreads();

  // epilogue: C/D layout VGPR r -> row (w*16 + r + 8*hi), col (nt*16 + l15)
  #pragma unroll
  for (int r = 0; r < 8; ++r) {
    int qrow = w * 16 + r + 8 * hi;
    float s = invq[qrow];
    size_t orow = (qBase + qrow) * NWAY;
    out[orow +  0 + l15] = acc0[r] * s;
    out[orow + 16 + l15] = acc1[r] * s;
    out[orow + 32 + l15] = acc2[r] * s;
    out[orow + 48 + l15] = acc3[r] * s;
  }
}

// ---------------------------------------------------------------------------
extern "C" void kernel_launch(void* const* d_in, const int* in_sizes, int n_in,
                              void* d_out, int out_size, void* d_ws, size_t ws_size,
                              hipStream_t stream) {
  const float* sup    = (const float*)d_in[0];
  const int*   slab   = (const int*)d_in[1];
  const float* qry    = (const float*)d_in[2];
  // d_in[3] = query_labels (unused by the reference output)
  const float* scaleP = (const float*)d_in[4];
  float* out = (float*)d_out;

  // workspace: [0, 8MB) partial sums (128 blocks x 64 classes x 256 dims),
  //            then 128KB normalized prototypes
  float* pSums = (float*)d_ws;
  float* protN = (float*)((char*)d_ws + (size_t)128 * NWAY * 256 * sizeof(float));

  proto_partial<<<128, 256, 0, stream>>>(sup, slab, pSums);
  proto_final  <<<NWAY, 256, 0, stream>>>(pSums, slab, scaleP, protN);
  cosine_gemm  <<<NQ / MBLK, 256, 0, stream>>>(qry, protN, out);
}